// GatedElmanCell_22823456211351
// MI455X (gfx1250) — compile-verified
//
#include <hip/hip_runtime.h>
#include <cmath>

// GatedElmanCell for MI455X (gfx1250), wave32, WMMA f32 16x16x4.
// Pipeline: init (zero barriers, copy h0) -> power-iteration (spectral scale)
//        -> batched input GEMMs (x@W_x^T, sigmoid(x@W_delta^T)) via WMMA
//        -> persistent scan kernel: W_h slice in LDS, per-step WMMA recurrence,
//           fused gate epilogue, software grid barrier per timestep.
//
// K-chunking trick: contraction order over K is free, so chunk c (8 K values)
// assigns lane-half 0 the float4 at k=8c and lane-half 1 the float4 at k=8c+4;
// .xy feeds WMMA #1 (k in {8c,8c+1,8c+4,8c+5}) and .zw feeds WMMA #2
// (k in {8c+2,8c+3,8c+6,8c+7}). A and B use the same permutation -> exact,
// and all fragment traffic is b128. Chunks are pipelined with a ping-pong
// (unroll-by-2) buffer scheme so no register-rotation movs appear in the loop.

typedef __attribute__((ext_vector_type(2))) float v2f;
typedef __attribute__((ext_vector_type(8))) float v8f;

static constexpr int Tn = 2048;
static constexpr int Bn = 16;
static constexpr int Dn = 1024;
static constexpr int SCAN_BLOCKS = Dn / 16;   // 64 feature tiles
static constexpr int NCHUNK = Dn / 8;         // 128 K-chunks of 8 (even)

__device__ __forceinline__ v8f wmma_f32_4(v2f a, v2f b, v8f c) {
#if defined(__gfx1250__) && __has_builtin(__builtin_amdgcn_wmma_f32_16x16x4_f32)
  // (neg_a, A, neg_b, B, c_mod, C, reuse_a, reuse_b)
  return __builtin_amdgcn_wmma_f32_16x16x4_f32(false, a, false, b, (short)0, c,
                                               false, false);
#else
  return c;  // fallback marker: histogram would show wmma==0
#endif
}

__device__ __forceinline__ v2f mk2(float x, float y) {
  v2f r; r.x = x; r.y = y; return r;
}

// ---------------------------------------------------------------------------
// Init: zero per-step barrier counters, copy h0 into h[0] of the output.
// ---------------------------------------------------------------------------
__global__ void gec_init(const float* __restrict__ h0, float* __restrict__ hbuf,
                         int* __restrict__ counters) {
  int i = blockIdx.x * blockDim.x + threadIdx.x;
  if (i < Tn) counters[i] = 0;
  if (i < Bn * Dn) hbuf[i] = h0[i];
}

// ---------------------------------------------------------------------------
// Spectral-norm power iteration (single 1024-thread block).
// u init is a deterministic hash (reference uses jax.random.normal(key(1))).
// ---------------------------------------------------------------------------
__device__ float block_reduce_sum(float val, float* red) {
  int t = threadIdx.x;
  red[t] = val;
  __syncthreads();
  for (int o = 512; o > 0; o >>= 1) {
    if (t < o) red[t] += red[t + o];
    __syncthreads();
  }
  float r = red[0];
  __syncthreads();
  return r;
}

__global__ __launch_bounds__(1024) void gec_power_iter(
    const float* __restrict__ Wh, float* __restrict__ scale_out) {
  __shared__ float u[Dn];
  __shared__ float v[Dn];
  __shared__ float red[Dn];
  const int t = threadIdx.x;

  unsigned s = (unsigned)t * 2654435761u + 12345u;
  s ^= s >> 16; s *= 2246822519u; s ^= s >> 13; s *= 3266489917u; s ^= s >> 16;
  u[t] = ((float)(s >> 8) * (1.0f / 16777216.0f)) * 2.0f - 1.0f;
  __syncthreads();

  float n = sqrtf(block_reduce_sum(u[t] * u[t], red));
  u[t] /= n;
  __syncthreads();

  for (int it = 0; it < 3; ++it) {
    float acc = 0.0f;  // v = W^T u
    for (int i = 0; i < Dn; ++i) acc += Wh[(size_t)i * Dn + t] * u[i];
    v[t] = acc;
    __syncthreads();
    n = sqrtf(block_reduce_sum(v[t] * v[t], red));
    v[t] /= (n + 1e-8f);
    __syncthreads();

    acc = 0.0f;  // u = W v
    for (int j = 0; j < Dn; ++j) acc += Wh[(size_t)t * Dn + j] * v[j];
    u[t] = acc;
    __syncthreads();
    n = sqrtf(block_reduce_sum(u[t] * u[t], red));
    u[t] /= (n + 1e-8f);
    __syncthreads();
  }

  float acc = 0.0f;  // sigma = |u^T (W v)|
  for (int j = 0; j < Dn; ++j) acc += Wh[(size_t)t * Dn + j] * v[j];
  float sigma = block_reduce_sum(u[t] * acc, red);
  if (t == 0) scale_out[0] = 0.99f / (fabsf(sigma) + 1e-8f);
}

// ---------------------------------------------------------------------------
// Batched input GEMMs: out[m,n] = sum_k x[m,k]*W[n,k] + bias[n]
// M = T*B = 32768, N = D = 1024, K = D. One wave per block computes a
// 16x64 output tile (4 WMMA accumulators). z==1 also applies sigmoid (delta).
// Grid: (colgrp=16, rowtile=2048, which=2); x-fastest order keeps the shared
// x row-tile hot in L2 while W_x/W_delta (8MB) stay fully L2-resident.
// ---------------------------------------------------------------------------
__global__ __launch_bounds__(32) void gec_input_gemm(
    const float* __restrict__ x, const float* __restrict__ Wx,
    const float* __restrict__ Wd, const float* __restrict__ b,
    const float* __restrict__ b_delta, float* __restrict__ cand,
    float* __restrict__ delta) {
  const int colgrp = blockIdx.x;          // 0..15 (4 col tiles each)
  const int rowt = blockIdx.y;            // 0..2047
  const int which = blockIdx.z;           // 0: cand/W_x  1: delta/W_delta
  const float* __restrict__ W = which ? Wd : Wx;
  const float* __restrict__ bias = which ? b_delta : b;
  float* __restrict__ out = which ? delta : cand;

  const int lane = threadIdx.x;
  const int half = lane >> 4;   // chunk base k = 8c + 4*half
  const int l = lane & 15;
  const int row0 = rowt * 16;

  v8f acc[4] = {{}, {}, {}, {}};
  const float* arow = x + (size_t)(row0 + l) * Dn + 4 * half;
  const float* brow[4];
#pragma unroll
  for (int ct = 0; ct < 4; ++ct) {
    const int col0 = (colgrp * 4 + ct) * 16;
    brow[ct] = W + (size_t)(col0 + l) * Dn + 4 * half;
  }

  // Ping-pong pipeline (unroll by 2): buffers alternate by code position,
  // no register rotation. A holds even chunks, B holds odd chunks.
  float4 aA, aB, bA[4], bB[4];
  aA = *(const float4*)(arow);
#pragma unroll
  for (int ct = 0; ct < 4; ++ct) bA[ct] = *(const float4*)(brow[ct]);

  for (int c = 0; c < NCHUNK - 2; c += 2) {
    aB = *(const float4*)(arow + 8 * (c + 1));
#pragma unroll
    for (int ct = 0; ct < 4; ++ct)
      bB[ct] = *(const float4*)(brow[ct] + 8 * (c + 1));
#pragma unroll
    for (int ct = 0; ct < 4; ++ct) {
      acc[ct] = wmma_f32_4(mk2(aA.x, aA.y), mk2(bA[ct].x, bA[ct].y), acc[ct]);
      acc[ct] = wmma_f32_4(mk2(aA.z, aA.w), mk2(bA[ct].z, bA[ct].w), acc[ct]);
    }
    aA = *(const float4*)(arow + 8 * (c + 2));
#pragma unroll
    for (int ct = 0; ct < 4; ++ct)
      bA[ct] = *(const float4*)(brow[ct] + 8 * (c + 2));
#pragma unroll
    for (int ct = 0; ct < 4; ++ct) {
      acc[ct] = wmma_f32_4(mk2(aB.x, aB.y), mk2(bB[ct].x, bB[ct].y), acc[ct]);
      acc[ct] = wmma_f32_4(mk2(aB.z, aB.w), mk2(bB[ct].z, bB[ct].w), acc[ct]);
    }
  }
  // Epilogue: chunk NCHUNK-2 already in A; load last chunk into B.
  aB = *(const float4*)(arow + 8 * (NCHUNK - 1));
#pragma unroll
  for (int ct = 0; ct < 4; ++ct)
    bB[ct] = *(const float4*)(brow[ct] + 8 * (NCHUNK - 1));
#pragma unroll
  for (int ct = 0; ct < 4; ++ct) {
    acc[ct] = wmma_f32_4(mk2(aA.x, aA.y), mk2(bA[ct].x, bA[ct].y), acc[ct]);
    acc[ct] = wmma_f32_4(mk2(aA.z, aA.w), mk2(bA[ct].z, bA[ct].w), acc[ct]);
    acc[ct] = wmma_f32_4(mk2(aB.x, aB.y), mk2(bB[ct].x, bB[ct].y), acc[ct]);
    acc[ct] = wmma_f32_4(mk2(aB.z, aB.w), mk2(bB[ct].z, bB[ct].w), acc[ct]);
  }

#pragma unroll
  for (int ct = 0; ct < 4; ++ct) {
    const int col0 = (colgrp * 4 + ct) * 16;
    const int n = col0 + l;
    const float bn = bias[n];
#pragma unroll
    for (int r = 0; r < 8; ++r) {
      const int m = r + half * 8;  // C layout: lanes16-31 hold M=8..15
      float val = acc[ct][r] + bn;
      if (which) val = 1.0f / (1.0f + expf(-val));  // pre-apply sigmoid
      out[(size_t)(row0 + m) * Dn + n] = val;
    }
  }
}

// ---------------------------------------------------------------------------
// Persistent scan: 64 blocks x 1 wave. Block j owns features [16j,16j+16),
// its W_h slice lives in LDS (64KB). Per step: 256 x wmma_f32_16x16x4
// (A = h_prev b128 from L2, B = ds_load_b128 from LDS), fused gate epilogue,
// then a release->acquire software grid barrier on counters[t].
// h state is written straight into d_out's h region ([T+1,B,D]).
// ---------------------------------------------------------------------------
__global__ __launch_bounds__(32) void gec_scan(
    const float* __restrict__ x, const float* __restrict__ Wh,
    const float* __restrict__ b_gate, const float* __restrict__ cand,
    const float* __restrict__ delta, const float* __restrict__ scale_p,
    float* __restrict__ outbuf, float* __restrict__ hbuf,
    int* __restrict__ counters, int nblocks) {
  __shared__ float wslice[16 * Dn];  // 64 KB: rows col0..col0+15 of W_h

  const int col0 = blockIdx.x * 16;
  const int lane = threadIdx.x;
  const int half = lane >> 4;
  const int l = lane & 15;

  // Stage this block's W_h slice into LDS once (float4 copies).
  const float4* src = (const float4*)(Wh + (size_t)col0 * Dn);
  float4* dst = (float4*)wslice;
  for (int i = lane; i < 16 * Dn / 4; i += 32) dst[i] = src[i];
  __syncthreads();

  const float scale = scale_p[0];
  const float bg = b_gate[col0 + l];
  const float* brow = wslice + l * Dn + 4 * half;

  for (int t = 0; t < Tn; ++t) {
    const float* __restrict__ h_prev = hbuf + (size_t)t * Bn * Dn;
    const float* arow = h_prev + (size_t)l * Dn + 4 * half;

    v8f acc = {};
    float4 aA = *(const float4*)(arow);
    float4 bA = *(const float4*)(brow);
    float4 aB, bB;
    for (int c = 0; c < NCHUNK - 2; c += 2) {
      aB = *(const float4*)(arow + 8 * (c + 1));
      bB = *(const float4*)(brow + 8 * (c + 1));
      acc = wmma_f32_4(mk2(aA.x, aA.y), mk2(bA.x, bA.y), acc);
      acc = wmma_f32_4(mk2(aA.z, aA.w), mk2(bA.z, bA.w), acc);
      aA = *(const float4*)(arow + 8 * (c + 2));
      bA = *(const float4*)(brow + 8 * (c + 2));
      acc = wmma_f32_4(mk2(aB.x, aB.y), mk2(bB.x, bB.y), acc);
      acc = wmma_f32_4(mk2(aB.z, aB.w), mk2(bB.z, bB.w), acc);
    }
    aB = *(const float4*)(arow + 8 * (NCHUNK - 1));
    bB = *(const float4*)(brow + 8 * (NCHUNK - 1));
    acc = wmma_f32_4(mk2(aA.x, aA.y), mk2(bA.x, bA.y), acc);
    acc = wmma_f32_4(mk2(aA.z, aA.w), mk2(bA.z, bA.w), acc);
    acc = wmma_f32_4(mk2(aB.x, aB.y), mk2(bB.x, bB.y), acc);
    acc = wmma_f32_4(mk2(aB.z, aB.w), mk2(bB.z, bB.w), acc);

#pragma unroll
    for (int r = 0; r < 8; ++r) {
      const int m = r + half * 8;           // batch index
      const int n = col0 + l;               // feature index
      const size_t idx = ((size_t)t * Bn + m) * Dn + n;
      const float dlt = delta[idx];         // sigmoid pre-applied
      const float hp = h_prev[(size_t)m * Dn + n];
      const float candv = tanhf(cand[idx] + acc[r] * scale);
      const float hn = (1.0f - dlt) * hp + dlt * candv;
      const float z = hn + x[idx] + bg;
      const float silu = z / (1.0f + expf(-z));
      hbuf[((size_t)(t + 1) * Bn + m) * Dn + n] = hn;
      outbuf[idx] = hn * silu;
    }

    // Grid-wide barrier for step t (all 64 waves are co-resident).
    __threadfence();  // release: make h[t+1] stores visible at agent scope
    if (lane == 0) {
      __hip_atomic_fetch_add(&counters[t], 1, __ATOMIC_ACQ_REL,
                             __HIP_MEMORY_SCOPE_AGENT);
      while (__hip_atomic_load(&counters[t], __ATOMIC_ACQUIRE,
                               __HIP_MEMORY_SCOPE_AGENT) < nblocks) {
        __builtin_amdgcn_s_sleep(2);
      }
    }
    __threadfence();  // acquire side for the whole wave
  }
}

// ---------------------------------------------------------------------------
extern "C" void kernel_launch(void* const* d_in, const int* in_sizes, int n_in,
                              void* d_out, int out_size, void* d_ws,
                              size_t ws_size, hipStream_t stream) {
  const float* x = (const float*)d_in[0];        // [T,B,D]
  const float* h0 = (const float*)d_in[1];       // [B,D]
  const float* W_x = (const float*)d_in[2];      // [D,D]
  const float* W_h = (const float*)d_in[3];      // [D,D]
  const float* W_delta = (const float*)d_in[4];  // [D,D]
  const float* b = (const float*)d_in[5];        // [D]
  const float* b_delta = (const float*)d_in[6];  // [D]
  const float* b_gate = (const float*)d_in[7];   // [D]

  float* outbuf = (float*)d_out;                       // [T,B,D]
  float* hbuf = outbuf + (size_t)Tn * Bn * Dn;         // [T+1,B,D]

  // Workspace layout
  char* ws = (char*)d_ws;
  float* scale = (float*)ws;                           // 1 float
  int* counters = (int*)(ws + 256);                    // Tn ints
  float* cand = (float*)(ws + 65536);                  // [T,B,D]
  float* delta = cand + (size_t)Tn * Bn * Dn;          // [T,B,D]

  gec_init<<<(Bn * Dn + 255) / 256, 256, 0, stream>>>(h0, hbuf, counters);
  gec_power_iter<<<1, 1024, 0, stream>>>(W_h, scale);
  gec_input_gemm<<<dim3(Dn / 64, (Tn * Bn) / 16, 2), 32, 0, stream>>>(
      x, W_x, W_delta, b, b_delta, cand, delta);
  gec_scan<<<SCAN_BLOCKS, 32, 0, stream>>>(x, W_h, b_gate, cand, delta, scale,
                                           outbuf, hbuf, counters, SCAN_BLOCKS);
}